// SimMIMLoss_19430432047575
// MI455X (gfx1250) — compile-verified
//
#include <hip/hip_runtime.h>
#include <hip/hip_bf16.h>

// ---------------------------------------------------------------------------
// SimMIM masked-normalized-L1 loss for MI455X (gfx1250).
// 47x47 box sums = banded-ones matmuls on V_WMMA_F32_16X16X4_F32 (full f32).
// Tile staging via GLOBAL_LOAD_ASYNC_TO_LDS_B128 (ASYNCcnt) inline asm.
// Memory-bound: ~78 MB mandatory HBM traffic -> ~3.3 us floor @ 23.3 TB/s;
// cross-tile re-reads (3.9x on input) are 192MB-L2 hits, not HBM.
// ---------------------------------------------------------------------------

typedef __attribute__((ext_vector_type(2))) float v2f;
typedef __attribute__((ext_vector_type(8))) float v8f;

#define IMG_H   224
#define IMG_W   224
#define PAD     23          // NORM_K // 2
#define TR      16          // output rows per tile (one WMMA M block)
#define LR      62          // rows loaded per tile (TR + 2*PAD)
#define LRP     64          // padded to 4 WMMA row blocks
#define TSTRIDE 276         // Traw row stride (dwords): 24 left pad + 224 + right pad
#define HSTRIDE 236         // Ht/Ht2 row stride (dwords)
#define NCB     14          // column blocks of 16 (224/16)
#define NRT     14          // row tiles (224/16)
#define NPLANES (64*3)
#define NWG     (NPLANES*NRT)
#define MASK_N  (64*3136)

static __device__ __forceinline__ int imin(int a, int b) { return a < b ? a : b; }
static __device__ __forceinline__ int imax(int a, int b) { return a > b ? a : b; }

static __device__ __forceinline__ v8f wmma4(v2f a, v2f b, v8f c) {
    // D(16x16,f32) = A(16x4,f32) * B(4x16,f32) + C  -- full-precision matrix path
    return __builtin_amdgcn_wmma_f32_16x16x4_f32(false, a, false, b, (short)0, c,
                                                 false, false);
}

// ---------------------------------------------------------------------------
// Kernel 0: zero the integer mask accumulator.
// ---------------------------------------------------------------------------
__global__ void simmim_init_kernel(int* msum) {
    if (threadIdx.x == 0) msum[0] = 0;
}

// ---------------------------------------------------------------------------
// Kernel 1: integer mask sum (int atomics are associative -> deterministic).
// ---------------------------------------------------------------------------
__global__ void simmim_mask_sum_kernel(const int* __restrict__ mask, int n,
                                       int* __restrict__ msum) {
    __shared__ int sred[256];
    int s = 0;
    for (int i = blockIdx.x * 256 + threadIdx.x; i < n; i += gridDim.x * 256)
        s += mask[i];
    sred[threadIdx.x] = s;
    __syncthreads();
    for (int st = 128; st > 0; st >>= 1) {
        if (threadIdx.x < st) sred[threadIdx.x] += sred[threadIdx.x + st];
        __syncthreads();
    }
    if (threadIdx.x == 0) atomicAdd(msum, sred[0]);
}

// ---------------------------------------------------------------------------
// Kernel 2: main fused kernel. One WG = 16 output rows x 224 cols of one
// (b,c) plane. Async-DMA stage -> WMMA banded matmuls -> finalize + reduce.
// ---------------------------------------------------------------------------
extern "C" __global__ void __launch_bounds__(256)
simmim_main_kernel(const float* __restrict__ img, const float* __restrict__ rec,
                   const int* __restrict__ mask, float* __restrict__ partials) {
    extern __shared__ float smem[];
    float* Traw = smem;                              // [LRP][TSTRIDE] col-padded raw t
    float* Ht   = smem + LRP * TSTRIDE;              // [LRP][HSTRIDE] horiz 47-sums of t
    float* Ht2  = Ht + LRP * HSTRIDE;                // [LRP][HSTRIDE] horiz 47-sums of t^2
    int*   mL   = (int*)(Ht2 + LRP * HSTRIDE);       // [4][56] mask patch rows
    float* red  = (float*)(mL + 224);                // [256] reduction scratch

    const int tid  = threadIdx.x;
    const int lane = tid & 31;
    const int wid  = tid >> 5;
    const int n    = lane & 15;     // N coord (B/C/D) and M coord (A) per WMMA layout
    const int half = lane >> 4;     // lane group 0-15 vs 16-31 (K offset +2)

    const int wg    = blockIdx.x;
    const int tileR = wg % NRT;
    const int plane = wg / NRT;     // b*3 + c
    const int b     = plane / 3;
    const int r0    = tileR * TR;

    const float* imgP = img + (size_t)plane * (IMG_H * IMG_W);
    const float* recP = rec + (size_t)plane * (IMG_H * IMG_W);

    // Prefetch recon tile early (global_prefetch_b8) to hide final-phase latency.
    for (int i = tid; i < TR * IMG_W; i += 256)
        __builtin_prefetch(recP + (size_t)(r0 + i / IMG_W) * IMG_W + (i % IMG_W), 0, 0);

    // Mask patch rows for this tile: patch rows r0/4 .. r0/4+3.
    {
        int pr0 = r0 >> 2;
        for (int i = tid; i < 4 * 56; i += 256)
            mL[i] = mask[(size_t)b * 3136 + (pr0 + i / 56) * 56 + (i % 56)];
    }

    // ---- Stage tile into LDS -----------------------------------------------
    // Interior (valid rows, cols [24,248)): GLOBAL_LOAD_ASYNC_TO_LDS_B128, GVS
    // mode (SGPR base + per-lane 32-bit offset). Zero-fill (margins, OOB rows,
    // pad rows) uses the DS pipe on strictly disjoint addresses, because async
    // and DS traffic are mutually unordered.
    for (int i = tid; i < LR * 56; i += 256) {
        int lr   = i / 56;
        int cq   = i % 56;
        int grow = r0 - PAD + lr;
        float* dst = Traw + lr * TSTRIDE + 24 + cq * 4;
        if (grow >= 0 && grow < IMG_H) {
            unsigned ldsA = (unsigned)(uintptr_t)dst;
            unsigned goff = (unsigned)((grow * IMG_W + cq * 4) * sizeof(float));
            asm volatile("global_load_async_to_lds_b128 %0, %1, %2 offset:0"
                         :: "v"(ldsA), "v"(goff), "s"(imgP) : "memory");
        } else {
            *(float4*)dst = make_float4(0.f, 0.f, 0.f, 0.f);
        }
    }
    // Zero column margins: 6 quads left ([0,24)) + 7 quads right ([248,276)).
    for (int i = tid; i < LRP * 13; i += 256) {
        int lr  = i / 13;
        int q   = i % 13;
        int col = (q < 6) ? (q * 4) : (248 + (q - 6) * 4);
        *(float4*)(Traw + lr * TSTRIDE + col) = make_float4(0.f, 0.f, 0.f, 0.f);
    }
    // Zero interiors of the two padding rows (62,63).
    for (int i = tid; i < 2 * 56; i += 256) {
        int lr = LR + i / 56;
        int cq = i % 56;
        *(float4*)(Traw + lr * TSTRIDE + 24 + cq * 4) = make_float4(0.f, 0.f, 0.f, 0.f);
    }
    asm volatile("s_wait_asynccnt 0x0" ::: "memory");
    __syncthreads();

    const float* TrawC = Traw + 24;   // logical column 0 of the image

    // ---- Horizontal pass: Ht = T x Band, Ht2 = T.^2 x Band -----------------
    // 4 row-blocks x 14 col-blocks = 56 WMMA output tiles, round-robin by wave.
    // A addresses step by +4B per K -> folded into ds_load immediate offsets.
    for (int t = wid; t < 4 * NCB; t += 8) {
        int mb   = t / NCB;
        int nb   = t % NCB;
        int rowA = mb * 16 + n;          // A: M = lane&15
        const float* aBase = TrawC + rowA * TSTRIDE + (nb * 16 - PAD) + half * 2;
        v8f acc  = {};
        v8f acc2 = {};
#pragma unroll
        for (int k0 = 0; k0 < 64; k0 += 4) {
            int kk = k0 + half * 2;      // K of VGPR0 component (VGPR1 = kk+1)
            float a0 = aBase[k0];
            float a1 = aBase[k0 + 1];
            // in-band iff 0 <= kk-n <= 46 (zero margins handle column OOB)
            float b0 = ((unsigned)(kk - n)     <= 46u) ? 1.f : 0.f;
            float b1 = ((unsigned)(kk + 1 - n) <= 46u) ? 1.f : 0.f;
            v2f A  = {a0, a1};
            v2f A2 = {a0 * a0, a1 * a1};
            v2f B  = {b0, b1};
            acc  = wmma4(A,  B, acc);
            acc2 = wmma4(A2, B, acc2);
        }
#pragma unroll
        for (int v = 0; v < 8; v++) {    // D: M = v + half*8, N = lane&15
            int row = mb * 16 + v + half * 8;
            int col = nb * 16 + n;
            Ht [row * HSTRIDE + col] = acc[v];
            Ht2[row * HSTRIDE + col] = acc2[v];
        }
    }
    __syncthreads();

    // ---- Vertical pass + finalize: S = Band x Ht, S2 = Band x Ht2 ----------
    float lsum = 0.f;
    for (int nb = wid; nb < NCB; nb += 8) {
        int colL = nb * 16 + n;
        const float* hBase  = Ht  + half * 2 * HSTRIDE + colL;
        const float* h2Base = Ht2 + half * 2 * HSTRIDE + colL;
        v8f acc  = {};
        v8f acc2 = {};
#pragma unroll
        for (int k0 = 0; k0 < 64; k0 += 4) {
            int kk = k0 + half * 2;
            // A band row M = lane&15 (= n): window covers loaded rows [M, M+46].
            float a0 = ((unsigned)(kk - n)     <= 46u) ? 1.f : 0.f;
            float a1 = ((unsigned)(kk + 1 - n) <= 46u) ? 1.f : 0.f;
            float b0 = hBase [ k0      * HSTRIDE];
            float b1 = hBase [(k0 + 1) * HSTRIDE];
            float c0 = h2Base[ k0      * HSTRIDE];
            float c1 = h2Base[(k0 + 1) * HSTRIDE];
            v2f A  = {a0, a1};
            v2f B1 = {b0, b1};
            v2f B2 = {c0, c1};
            acc  = wmma4(A, B1, acc);
            acc2 = wmma4(A, B2, acc2);
        }
        int gcol   = nb * 16 + n;
        float cntX = (float)(imin(gcol + PAD, IMG_W - 1) - imax(gcol - PAD, 0) + 1);
#pragma unroll
        for (int v = 0; v < 8; v++) {
            int M    = v + half * 8;     // D layout: M = v + half*8
            int grow = r0 + M;
            float cntY = (float)(imin(grow + PAD, IMG_H - 1) - imax(grow - PAD, 0) + 1);
            float cnt  = cntX * cntY;
            float s1 = acc[v], s2 = acc2[v];
            float mean = s1 / cnt;
            float varv = (s2 / cnt - mean * mean) * (cnt / (cnt - 1.f));
            varv = fmaxf(varv, 0.f);
            float tval  = TrawC[(M + PAD) * TSTRIDE + gcol];
            float normt = (tval - mean) * rsqrtf(varv + 1e-6f);
            float rv = recP[(size_t)grow * IMG_W + gcol];
            float mv = (float)mL[(M >> 2) * 56 + (gcol >> 2)];
            lsum += fabsf(normt - rv) * mv;
        }
    }

    // Deterministic block tree reduce -> one partial per WG.
    red[tid] = lsum;
    __syncthreads();
    for (int s = 128; s > 0; s >>= 1) {
        if (tid < s) red[tid] += red[tid + s];
        __syncthreads();
    }
    if (tid == 0) partials[wg] = red[0];
}

// ---------------------------------------------------------------------------
// Kernel 3: fixed-order final reduction + scalar epilogue.
// ---------------------------------------------------------------------------
__global__ void simmim_final_kernel(const float* __restrict__ partials,
                                    const int* __restrict__ msum,
                                    float* __restrict__ out) {
    __shared__ float sred[256];
    float s = 0.f;
    for (int i = threadIdx.x; i < NWG; i += 256) s += partials[i];
    sred[threadIdx.x] = s;
    __syncthreads();
    for (int st = 128; st > 0; st >>= 1) {
        if (threadIdx.x < st) sred[threadIdx.x] += sred[threadIdx.x + st];
        __syncthreads();
    }
    if (threadIdx.x == 0) {
        float num   = sred[0];
        float msumf = 16.f * (float)msum[0];            // m.sum() (mask repeated 4x4)
        float den   = msumf * 3.f + 1e-5f;              // m.sum()*C + 1e-5
        out[0] = num / den / 3.f;                       // / C
    }
}

// ---------------------------------------------------------------------------
// Host entry point.
// ---------------------------------------------------------------------------
extern "C" void kernel_launch(void* const* d_in, const int* in_sizes, int n_in,
                              void* d_out, int out_size, void* d_ws, size_t ws_size,
                              hipStream_t stream) {
    const float* img  = (const float*)d_in[0];
    const float* rec  = (const float*)d_in[1];
    const int*   mask = (const int*)d_in[2];
    float*       out  = (float*)d_out;

    float* partials = (float*)d_ws;                              // NWG floats
    int*   msum     = (int*)((char*)d_ws + NWG * sizeof(float)); // 2688*4 = 10752 B in

    const size_t lds_bytes =
        (size_t)(LRP * TSTRIDE + 2 * LRP * HSTRIDE + 224 + 256) * 4;  // ~193 KB of 320 KB
    (void)hipFuncSetAttribute((const void*)simmim_main_kernel,
                              hipFuncAttributeMaxDynamicSharedMemorySize,
                              (int)lds_bytes);

    simmim_init_kernel<<<1, 32, 0, stream>>>(msum);
    simmim_mask_sum_kernel<<<392, 256, 0, stream>>>(mask, MASK_N, msum);
    simmim_main_kernel<<<NWG, 256, lds_bytes, stream>>>(img, rec, mask, partials);
    simmim_final_kernel<<<1, 256, 0, stream>>>(partials, msum, out);
}